// SoftConstrainedIPMModule_26250840113593
// MI455X (gfx1250) — compile-verified
//
#include <hip/hip_runtime.h>
#include <stdint.h>

// ---------------------------------------------------------------------------
// Batched soft-constrained IPM solver, MI455X (gfx1250).
// 32768 independent 12-var / 11-constraint QPs. One problem per lane (wave32).
// KKT system solved by block elimination -> tridiagonal Thomas solve (the
// dense 23x23 inverse in the reference is mathematically equivalent but ~25x
// the FLOPs and cannot fill a 16x16 WMMA tile with per-problem matrices).
// Inputs staged via CDNA5 async global->LDS DMA (ASYNCcnt path).
// Global max-residual reduced per-block in LDS (uint order == float order for
// x >= 0, NaN dominates like jnp.max), then one atomicMax per block.
// ---------------------------------------------------------------------------

#define NPROB 32768
#define NN    12
#define MM    11
#define WW    1024
#define BLK   256
#define EPS_IPM  0.01f
#define BETA1_C  0.5f
#define BETA2_C  0.055f

// ------------------------- Threefry-2x32 (JAX PRNG) ------------------------
__device__ __forceinline__ uint32_t rotl32(uint32_t v, int n) {
  return (v << n) | (v >> (32 - n));
}

__device__ __forceinline__ void threefry2x32(uint32_t k0, uint32_t k1,
                                             uint32_t c0, uint32_t c1,
                                             uint32_t &o0, uint32_t &o1) {
  uint32_t ks2 = k0 ^ k1 ^ 0x1BD11BDAu;
  uint32_t x0 = c0 + k0, x1 = c1 + k1;
  const int ra[4] = {13, 15, 26, 6};
  const int rb[4] = {17, 29, 16, 24};
#pragma unroll
  for (int i = 0; i < 4; i++) { x0 += x1; x1 = rotl32(x1, ra[i]); x1 ^= x0; }
  x0 += k1;  x1 += ks2 + 1u;
#pragma unroll
  for (int i = 0; i < 4; i++) { x0 += x1; x1 = rotl32(x1, rb[i]); x1 ^= x0; }
  x0 += ks2; x1 += k0 + 2u;
#pragma unroll
  for (int i = 0; i < 4; i++) { x0 += x1; x1 = rotl32(x1, ra[i]); x1 ^= x0; }
  x0 += k0;  x1 += k1 + 3u;
#pragma unroll
  for (int i = 0; i < 4; i++) { x0 += x1; x1 = rotl32(x1, rb[i]); x1 ^= x0; }
  x0 += k1;  x1 += ks2 + 4u;
#pragma unroll
  for (int i = 0; i < 4; i++) { x0 += x1; x1 = rotl32(x1, ra[i]); x1 ^= x0; }
  x0 += ks2; x1 += k0 + 5u;
  o0 = x0; o1 = x1;
}

// jax.random.uniform element e of a flat array of size 2*half (bit-exact):
// counts = iota(n); pair i = (i, i+half); bits = [y0 of pairs, y1 of pairs].
__device__ __forceinline__ float tf_uniform01(uint32_t k0, uint32_t k1,
                                              uint32_t e, uint32_t half) {
  uint32_t c0 = (e < half) ? e : (e - half);
  uint32_t c1 = (e < half) ? (e + half) : e;
  uint32_t y0, y1;
  threefry2x32(k0, k1, c0, c1, y0, y1);
  uint32_t bits = (e < half) ? y0 : y1;
  float u = __uint_as_float((bits >> 9) | 0x3F800000u) - 1.0f;
  return fmaxf(0.0f, u);
}

// ------------------------- residual norm (23-vector) -----------------------
__device__ __forceinline__ float resid_norm(const float diag[NN], const float bb[NN],
                                            const float S[NN], const float L[MM],
                                            float rt) {
  float acc = 0.0f;
#pragma unroll
  for (int n = 0; n < NN; n++) {
    float hs = diag[n] * S[n] - (n > 0 ? 4.0f * S[n - 1] : 0.0f);
    float at = (n < MM ? L[n] : 0.0f) - (n > 0 ? L[n - 1] : 0.0f);
    float r1 = hs + bb[n] + at;
    acc += r1 * r1;
  }
#pragma unroll
  for (int m = 0; m < MM; m++) {
    float g  = S[m] - S[m + 1];
    float r2 = -L[m] * g - rt;
    acc += r2 * r2;
  }
  return sqrtf(acc);
}

// ------------------------- init: S0 = sort(Mu), Lam0, beta3 ----------------
__global__ __launch_bounds__(BLK) void ipm_init(const float* __restrict__ Mu,
                                                float* __restrict__ Sws,
                                                float* __restrict__ Lws,
                                                float* __restrict__ b3ws,
                                                unsigned* __restrict__ flags) {
  int p = blockIdx.x * BLK + threadIdx.x;
  if (p == 0) {
    flags[0] = 0x7F800000u;  // +inf: iteration 1 always runs
#pragma unroll
    for (int j = 1; j <= 8; j++) flags[j] = 0u;
  }
  int b = p >> 10, w = p & 1023;

  float s[NN];
#pragma unroll
  for (int n = 0; n < NN; n++)
    s[n] = Mu[((size_t)(b * NN + n)) * WW + w];

  // ascending sort, fully unrolled compare-exchange (registers only)
#pragma unroll
  for (int i = 0; i < NN; i++) {
#pragma unroll
    for (int j = 0; j < NN - 1; j++) {
      float a = s[j], c = s[j + 1];
      s[j]     = fminf(a, c);
      s[j + 1] = fmaxf(a, c);
    }
  }

  // jax.random.split(key(42)): root key data = (0, 42)
  uint32_t a0, a1, b0, b1;
  threefry2x32(0u, 42u, 0u, 2u, a0, b0);  // pair (0,2): y0 -> k1[0], y1 -> k2[0]
  threefry2x32(0u, 42u, 1u, 3u, a1, b1);  // pair (1,3): y0 -> k1[1], y1 -> k2[1]

#pragma unroll
  for (int m = 0; m < MM; m++) {
    uint32_t e = (uint32_t)p * 11u + (uint32_t)m;     // flat (B,W,M,1), size 360448
    Lws[m * NPROB + p] = tf_uniform01(a0, a1, e, 180224u);
  }
  b3ws[p] = 10.0f + tf_uniform01(b0, b1, (uint32_t)p, 16384u);  // (B,W,1,1)

#pragma unroll
  for (int n = 0; n < NN; n++) Sws[n * NPROB + p] = s[n];
}

// ------------------------- one outer IPM iteration -------------------------
__global__ __launch_bounds__(BLK) void ipm_iter(const float* __restrict__ clp,
                                                const float* __restrict__ Mu,
                                                const float* __restrict__ Sg,
                                                const float* __restrict__ Rr,
                                                float* __restrict__ Sws,
                                                float* __restrict__ Lws,
                                                const float* __restrict__ b3ws,
                                                unsigned* __restrict__ flags,
                                                int k) {
  // Global stopping rule of the reference: run while max-over-batch R2Norm >= eps.
  if (k > 1) {
    float prevmax = __uint_as_float(flags[k - 1]);
    if (prevmax < EPS_IPM) return;      // uniform across grid
  }

  __shared__ float    sMu[NN * BLK];
  __shared__ float    sSg[NN * BLK];
  __shared__ float    sR [NN * BLK];
  __shared__ unsigned red[BLK];

  const int t  = threadIdx.x;
  const int p0 = blockIdx.x * BLK;      // 256 consecutive w within one b
  const int b  = p0 >> 10;
  const int wb = p0 & 1023;

  // CDNA5 async DMA: global -> LDS, 9 x B128 per lane covers 3 x 12KB tiles.
  {
    const float* srcs[3] = {Mu, Sg, Rr};
    float*       dsts[3] = {sMu, sSg, sR};
#pragma unroll
    for (int r3 = 0; r3 < 3; r3++) {
      const float* src = srcs[r3] + ((size_t)(b * NN)) * WW + wb;
      float*       dst = dsts[r3];
#pragma unroll
      for (int j = 0; j < 3; j++) {
        int e   = j * 1024 + t * 4;     // element index in 12x256 tile
        int n   = e >> 8;
        int col = e & 255;
        uint32_t lw = (uint32_t)(uintptr_t)(dst + e);               // LDS byte addr
        uint64_t ga = (uint64_t)(uintptr_t)(src + (size_t)n * WW + col);
        asm volatile("global_load_async_to_lds_b128 %0, %1, off"
                     :: "v"(lw), "v"(ga) : "memory");
      }
    }
    asm volatile("s_wait_asynccnt 0" ::: "memory");
    __syncthreads();
  }

  const int p  = p0 + t;
  const float cl = clp[0];

  // Build H diagonal and b from staged inputs.
  float diag[NN], bb[NN];
  {
    float rr[NN];
#pragma unroll
    for (int n = 0; n < NN; n++) rr[n] = sR[n * BLK + t];
#pragma unroll
    for (int n = 0; n < NN; n++) {
      float mu = sMu[n * BLK + t];
      float sg = sSg[n * BLK + t];
      diag[n] = cl / sg + 2.0f + (n < NN - 1 ? 2.0f : 0.0f);
      bb[n]   = -cl * mu / sg - 2.0f * rr[n] + (n < NN - 1 ? 2.0f * rr[n + 1] : 0.0f);
    }
  }

  // Load state (S0, Lam0 of this outer iteration).
  float S[NN], L[MM];
#pragma unroll
  for (int n = 0; n < NN; n++) S[n] = Sws[n * NPROB + p];
#pragma unroll
  for (int m = 0; m < MM; m++) L[m] = Lws[m * NPROB + p];
  const float beta3 = b3ws[p];

  // ---- t and residual at (S, Lam) ----
  float g[MM], dotv = 0.0f;
#pragma unroll
  for (int m = 0; m < MM; m++) { g[m] = S[m] - S[m + 1]; dotv += g[m] * L[m]; }
  const float tq = -beta3 * (float)MM / dotv;
  const float rt = 1.0f / tq;

  float R1[NN], R2[MM], rnsq = 0.0f;
#pragma unroll
  for (int n = 0; n < NN; n++) {
    float hs = diag[n] * S[n] - (n > 0 ? 4.0f * S[n - 1] : 0.0f);
    float at = (n < MM ? L[n] : 0.0f) - (n > 0 ? L[n - 1] : 0.0f);
    R1[n] = hs + bb[n] + at;
    rnsq += R1[n] * R1[n];
  }
#pragma unroll
  for (int m = 0; m < MM; m++) {
    R2[m] = -L[m] * g[m] - rt;
    rnsq += R2[m] * R2[m];
  }
  const float RNorm = sqrtf(rnsq);

  // ---- KKT solve via Schur complement: tridiagonal Thomas (no pivoting:
  //      lambda>0, g<=0 => diagonally dominant) ----
  float wv[MM];
#pragma unroll
  for (int m = 0; m < MM; m++) wv[m] = L[m] / g[m];

  float cp[NN], dp[NN];
  {
    float mj = diag[0] - wv[0];
    cp[0] = wv[0] / mj;
    dp[0] = (-R1[0] - R2[0] / g[0]) / mj;
  }
#pragma unroll
  for (int j = 1; j < NN; j++) {
    float lo  = -4.0f + wv[j - 1];
    float mn  = diag[j] - (j < MM ? wv[j] : 0.0f) - wv[j - 1];
    float up  = (j < MM ? wv[j] : 0.0f);
    float rh  = -R1[j] - (j < MM ? R2[j] / g[j] : 0.0f) + R2[j - 1] / g[j - 1];
    float den = mn - lo * cp[j - 1];
    cp[j] = up / den;
    dp[j] = (rh - lo * dp[j - 1]) / den;
  }
  float dS[NN];
  dS[NN - 1] = dp[NN - 1];
#pragma unroll
  for (int j = NN - 2; j >= 0; j--) dS[j] = dp[j] - cp[j] * dS[j + 1];

  float dl[MM];
#pragma unroll
  for (int m = 0; m < MM; m++)
    dl[m] = (R2[m] - L[m] * (dS[m] - dS[m + 1])) / g[m];

  // ---- step length: alpha = 0.99 * min_m(-lam/negPDL), NaN -> 1 ----
  float alpha = 1e30f;
#pragma unroll
  for (int m = 0; m < MM; m++) {
    float den = (dl[m] < 0.0f) ? dl[m] : -L[m];
    float am  = -L[m] / den;
    if (!(am == am)) am = 1.0f;
    alpha = fminf(alpha, am);
  }
  alpha *= 0.99f;

  // ---- feasibility backtracking: while any (A S_new > d) halve alpha ----
  float Sn[NN];
  for (;;) {
#pragma unroll
    for (int n = 0; n < NN; n++) Sn[n] = S[n] + alpha * dS[n];
    bool viol = false;
#pragma unroll
    for (int m = 0; m < MM; m++) viol = viol || ((Sn[m] - Sn[m + 1]) > 0.0f);
    if (!viol) break;
    alpha *= BETA1_C;
  }

  // ---- merit line search on ||R(S_new, Lam_new, t)|| ----
  float Ln[MM];
#pragma unroll
  for (int m = 0; m < MM; m++) Ln[m] = L[m] + alpha * dl[m];
  float r2n = resid_norm(diag, bb, Sn, Ln, rt);
  while (r2n > (1.0f - BETA2_C * alpha) * RNorm) {
    alpha *= BETA1_C;
#pragma unroll
    for (int n = 0; n < NN; n++) Sn[n] = S[n] + alpha * dS[n];
#pragma unroll
    for (int m = 0; m < MM; m++) Ln[m] = L[m] + alpha * dl[m];
    r2n = resid_norm(diag, bb, Sn, Ln, rt);
  }

  // ---- commit state ----
#pragma unroll
  for (int n = 0; n < NN; n++) Sws[n * NPROB + p] = Sn[n];
#pragma unroll
  for (int m = 0; m < MM; m++) Lws[m * NPROB + p] = Ln[m];

  // ---- block-level max reduction on float bits (x>=0: uint order == float
  //      order; NaN bits > +inf bits, so NaN propagates like jnp.max), then a
  //      single atomic per block instead of 32768 atomics on one address. ----
  red[t] = __float_as_uint(r2n);
  __syncthreads();
#pragma unroll
  for (int s = BLK / 2; s > 0; s >>= 1) {
    if (t < s) {
      unsigned x = red[t], y = red[t + s];
      red[t] = (x > y) ? x : y;
    }
    __syncthreads();
  }
  if (t == 0) atomicMax(&flags[k], red[0]);
}

// ------------------------- writeout: (p,n) -> (B,N,W) ----------------------
__global__ __launch_bounds__(BLK) void ipm_writeout(const float* __restrict__ Sws,
                                                    float* __restrict__ out) {
  int idx = blockIdx.x * BLK + threadIdx.x;           // 0 .. 12*NPROB-1
  int w    = idx & 1023;
  int rest = idx >> 10;
  int n    = rest % NN;
  int b    = rest / NN;
  int p    = (b << 10) | w;
  out[idx] = Sws[n * NPROB + p];
}

// ---------------------------------------------------------------------------
extern "C" void kernel_launch(void* const* d_in, const int* in_sizes, int n_in,
                              void* d_out, int out_size, void* d_ws, size_t ws_size,
                              hipStream_t stream) {
  const float* cl = (const float*)d_in[0];
  const float* Mu = (const float*)d_in[1];
  const float* Sg = (const float*)d_in[2];
  const float* Rr = (const float*)d_in[3];
  float* out = (float*)d_out;

  // workspace layout: S[12*NP] | Lam[11*NP] | beta3[NP] | flags[9]  (~3.1 MB)
  float* Sws  = (float*)d_ws;
  float* Lws  = Sws + (size_t)NN * NPROB;
  float* b3ws = Lws + (size_t)MM * NPROB;
  unsigned* flags = (unsigned*)(b3ws + NPROB);

  ipm_init<<<NPROB / BLK, BLK, 0, stream>>>(Mu, Sws, Lws, b3ws, flags);
  for (int k = 1; k <= 8; k++)  // MAX_ITERS=7 -> at most 8 outer bodies
    ipm_iter<<<NPROB / BLK, BLK, 0, stream>>>(cl, Mu, Sg, Rr, Sws, Lws, b3ws, flags, k);
  ipm_writeout<<<(NN * NPROB) / BLK, BLK, 0, stream>>>(Sws, out);
}